// GlmLayer_80968723464473
// MI455X (gfx1250) — compile-verified
//
#include <hip/hip_runtime.h>
#include <hip/hip_bf16.h>
#include <math.h>

typedef float v2f __attribute__((ext_vector_type(2)));
typedef float v8f __attribute__((ext_vector_type(8)));
typedef int   v4i __attribute__((vector_size(16)));

// ---- problem constants (from reference) ----
constexpr int B_   = 64;
constexpr int S_   = 1024;
constexpr int NH_  = 16;
constexpr int DKN_ = 128;
constexpr int DKR_ = 64;
constexpr int DV_  = 128;
constexpr int QHD_ = DKN_ + DKR_;          // 192
constexpr int HD_  = 2048;
constexpr int E_   = 32;
constexpr int I_   = 1408;
constexpr int TOPK_ = 4;
constexpr float EPS_ = 1e-6f;

// ---- tiling ----
constexpr int MT = 64;      // always == B_
constexpr int NT = 64;
constexpr int KC = 32;      // K chunk (8 WMMA k-steps of 4)
constexpr int AP  = KC + 4; // A / NT-B LDS pitch (36 floats = 144B, 16B aligned, conflict-free)
constexpr int BPN = NT + 4; // NN-B LDS pitch (68 floats = 272B)

// =====================================================================
// Async global -> LDS copy (16B per lane): GLOBAL_LOAD_ASYNC_TO_LDS_B128
// (ASYNCcnt). Builtin expects (AS1 v4i*, AS3 v4i*, imm offset, imm cpol).
// =====================================================================
typedef __attribute__((address_space(1))) v4i GV4i;
typedef __attribute__((address_space(3))) v4i LV4i;

#if __has_builtin(__builtin_amdgcn_global_load_async_to_lds_b128) && \
    __has_builtin(__builtin_amdgcn_s_wait_asynccnt)
#define ASYNC_MODE 1
#elif defined(__gfx1250__)
#define ASYNC_MODE 2
#else
#define ASYNC_MODE 0
#endif

__device__ __forceinline__ void cp16(float* dst, const float* src)
{
#if ASYNC_MODE == 1
    __builtin_amdgcn_global_load_async_to_lds_b128((GV4i*)src, (LV4i*)dst, 0, 0);
#elif ASYNC_MODE == 2
    unsigned int ldsa = (unsigned int)(unsigned long long)dst; // LDS aperture: addr[31:0]
    asm volatile("global_load_async_to_lds_b128 %0, %1, off"
                 :: "v"(ldsa), "v"(src) : "memory");
#else
    *(float4*)dst = *(const float4*)src;
#endif
}

__device__ __forceinline__ void cp_wait()
{
#if ASYNC_MODE == 1
    __builtin_amdgcn_s_wait_asynccnt(0);
#elif ASYNC_MODE == 2
    asm volatile("s_wait_asynccnt 0x0" ::: "memory");
#endif
}

// =====================================================================
// Tile loaders: 256 threads, each copies 16B chunks.
// A tile: [64][KC] from row-major A (lda), dst pitch AP.
// NN  B : [KC][NT] from row-major B[k][n] (ldb), dst pitch BPN.
// NT  B : kept [N][K] in LDS (pitch AP) -- contiguous B128 both sides.
// =====================================================================
__device__ __forceinline__ void loadA(float* As, const float* __restrict__ A,
                                      long lda, int k0)
{
#pragma unroll
    for (int j = 0; j < 2; ++j) {
        int idx = threadIdx.x + j * 256;           // 512 float4s
        int r = idx >> 3, c4 = (idx & 7) * 4;
        cp16(As + r * AP + c4, A + (size_t)r * lda + k0 + c4);
    }
}
__device__ __forceinline__ void loadB_nn(float* Bs, const float* __restrict__ Bm,
                                         long ldb, int n0, int k0)
{
#pragma unroll
    for (int j = 0; j < 2; ++j) {
        int idx = threadIdx.x + j * 256;
        int kk = idx >> 4, nn4 = (idx & 15) * 4;
        cp16(Bs + kk * BPN + nn4, Bm + (size_t)(k0 + kk) * ldb + n0 + nn4);
    }
}
__device__ __forceinline__ void loadB_nt(float* BT, const float* __restrict__ Bm,
                                         long ldb, int n0, int k0)
{
#pragma unroll
    for (int j = 0; j < 2; ++j) {
        int idx = threadIdx.x + j * 256;
        int nn = idx >> 3, kk4 = (idx & 7) * 4;
        cp16(BT + nn * AP + kk4, Bm + (size_t)(n0 + nn) * ldb + k0 + kk4);
    }
}

// =====================================================================
// WMMA fragment math (V_WMMA_F32_16X16X4_F32, wave32)
// A frag (16x4): lanes 0-15 hold K={k0,k0+1}, lanes 16-31 K={k0+2,k0+3}
// C/D: VGPR r -> M=r (lanes 0-15), M=r+8 (lanes 16-31); N = lane&15
// =====================================================================
__device__ __forceinline__ v8f mma_nn(const float* As, const float* Bs,
                                      int mfrag, int nfrag, v8f acc)
{
    const int lane = threadIdx.x & 31;
    const int half = lane >> 4, l = lane & 15;
    const float* arow = As + (mfrag * 16 + l) * AP;
    const float* bcol = Bs + (nfrag * 16 + l);
#pragma unroll
    for (int k0 = 0; k0 < KC; k0 += 4) {
        v2f a, b;
        a.x = arow[k0 + 2 * half + 0];
        a.y = arow[k0 + 2 * half + 1];
        b.x = bcol[(k0 + 2 * half + 0) * BPN];
        b.y = bcol[(k0 + 2 * half + 1) * BPN];
        acc = __builtin_amdgcn_wmma_f32_16x16x4_f32(false, a, false, b,
                                                    (short)0, acc, false, false);
    }
    return acc;
}
__device__ __forceinline__ v8f mma_nt(const float* As, const float* BT,
                                      int mfrag, int nfrag, v8f acc)
{
    const int lane = threadIdx.x & 31;
    const int half = lane >> 4, l = lane & 15;
    const float* arow = As + (mfrag * 16 + l) * AP;
    const float* brow = BT + (nfrag * 16 + l) * AP;   // B kept [n][k]
#pragma unroll
    for (int k0 = 0; k0 < KC; k0 += 4) {
        v2f a, b;
        a.x = arow[k0 + 2 * half + 0];
        a.y = arow[k0 + 2 * half + 1];
        b.x = brow[k0 + 2 * half + 0];
        b.y = brow[k0 + 2 * half + 1];
        acc = __builtin_amdgcn_wmma_f32_16x16x4_f32(false, a, false, b,
                                                    (short)0, acc, false, false);
    }
    return acc;
}

__device__ __forceinline__ void store_tile(float* __restrict__ C, int ldc, int n0,
                                           int mfrag, int nfrag, v8f acc,
                                           const float* __restrict__ res, int ldr)
{
    const int lane = threadIdx.x & 31;
    const int half = lane >> 4;
    const int n = n0 + nfrag * 16 + (lane & 15);
#pragma unroll
    for (int r = 0; r < 8; ++r) {
        const int m = mfrag * 16 + half * 8 + r;
        float v = acc[r];
        if (res) v += res[(size_t)m * ldr + n];
        C[(size_t)m * ldc + n] = v;
    }
}

// =====================================================================
// Generic NN GEMM: C[64,N] = A[64,K] @ B[K,N]  (+ optional residual)
// =====================================================================
__global__ __launch_bounds__(256)
void gemm64_kernel(const float* __restrict__ A, int lda,
                   const float* __restrict__ Bm, long ldb,
                   float* __restrict__ C, int ldc,
                   const float* __restrict__ res, int ldr, int K)
{
    __shared__ __align__(16) float As[MT * AP];
    __shared__ __align__(16) float Bs[KC * BPN];
    const int n0 = blockIdx.x * NT;
    const int wave = threadIdx.x >> 5;
    const int u0 = wave, u1 = wave + 8;          // 16 (mfrag,nfrag) units / 8 waves
    v8f acc0 = {}; v8f acc1 = {};

    for (int k0 = 0; k0 < K; k0 += KC) {
        __syncthreads();
        loadA(As, A, lda, k0);
        loadB_nn(Bs, Bm, ldb, n0, k0);
        cp_wait();
        __syncthreads();
        acc0 = mma_nn(As, Bs, u0 >> 2, u0 & 3, acc0);
        acc1 = mma_nn(As, Bs, u1 >> 2, u1 & 3, acc1);
    }
    store_tile(C, ldc, n0, u0 >> 2, u0 & 3, acc0, res, ldr);
    store_tile(C, ldc, n0, u1 >> 2, u1 & 3, acc1, res, ldr);
}

// =====================================================================
// RMSNorm
// =====================================================================
__global__ __launch_bounds__(256)
void rmsnorm_kernel(const float* __restrict__ x, const float* __restrict__ w,
                    float* __restrict__ y)
{
    const int b = blockIdx.x;
    const float* xr = x + (size_t)b * HD_;
    __shared__ float red[8];
    float s = 0.f;
    for (int i = threadIdx.x; i < HD_; i += 256) { float v = xr[i]; s += v * v; }
    for (int off = 16; off; off >>= 1) s += __shfl_xor(s, off, 32);
    if ((threadIdx.x & 31) == 0) red[threadIdx.x >> 5] = s;
    __syncthreads();
    if (threadIdx.x == 0) {
        float t = 0.f;
        for (int i = 0; i < 8; ++i) t += red[i];
        red[0] = rsqrtf(t / (float)HD_ + EPS_);
    }
    __syncthreads();
    const float inv = red[0];
    for (int i = threadIdx.x; i < HD_; i += 256)
        y[(size_t)b * HD_ + i] = xr[i] * inv * w[i];
}

// =====================================================================
// Flash-decode attention; new token's k/v substituted at slot_mapping[b]
// (caches are never mutated).
// =====================================================================
__global__ __launch_bounds__(128)
void attn_kernel(const float* __restrict__ qf,      // [B, NH*QHD]
                 const float* __restrict__ kvf,     // [B, NH*(DKN+DV)+DKR]
                 const float* __restrict__ k_cache, // [B,S,NH,DKN]
                 const float* __restrict__ v_cache, // [B,S,NH,DV]
                 const int* __restrict__ slot_mapping,
                 const int* __restrict__ seq_lens,
                 float* __restrict__ attn_out)      // [B, NH*DV]
{
    const int h = blockIdx.x, b = blockIdx.y;
    const int lane = threadIdx.x & 31, wv = threadIdx.x >> 5;
    const int len = seq_lens[b], slot = slot_mapping[b];
    const float scale = rsqrtf((float)QHD_);

    const size_t kvrow = (size_t)b * (NH_ * (DKN_ + DV_) + DKR_);
    const float4 q4 = *(const float4*)(qf + (size_t)b * (NH_ * QHD_) + h * QHD_ + lane * 4);
    const float* newk = kvf + kvrow + h * DKN_;
    const float* newv = kvf + kvrow + NH_ * DKN_ + h * DV_;

    float m = -INFINITY, lsum = 0.f;
    float4 av = {0.f, 0.f, 0.f, 0.f};
    for (int s = wv; s < len; s += 4) {
        const float* kr = (s == slot) ? newk
            : (k_cache + ((size_t)b * S_ * NH_ + (size_t)s * NH_ + h) * DKN_);
        float4 k4 = *(const float4*)(kr + lane * 4);
        float d = q4.x * k4.x + q4.y * k4.y + q4.z * k4.z + q4.w * k4.w;
        for (int off = 16; off; off >>= 1) d += __shfl_xor(d, off, 32);
        d *= scale;
        const float mn = fmaxf(m, d);
        const float corr = __expf(m - mn);
        const float p = __expf(d - mn);
        const float* vr = (s == slot) ? newv
            : (v_cache + ((size_t)b * S_ * NH_ + (size_t)s * NH_ + h) * DV_);
        float4 v4 = *(const float4*)(vr + lane * 4);
        av.x = av.x * corr + p * v4.x;
        av.y = av.y * corr + p * v4.y;
        av.z = av.z * corr + p * v4.z;
        av.w = av.w * corr + p * v4.w;
        lsum = lsum * corr + p;
        m = mn;
    }
    __shared__ float sm[4], sl[4];
    __shared__ float sacc[4][DV_];
    sacc[wv][lane * 4 + 0] = av.x; sacc[wv][lane * 4 + 1] = av.y;
    sacc[wv][lane * 4 + 2] = av.z; sacc[wv][lane * 4 + 3] = av.w;
    if (lane == 0) { sm[wv] = m; sl[wv] = lsum; }
    __syncthreads();
    if (wv == 0) {
        const float M = fmaxf(fmaxf(sm[0], sm[1]), fmaxf(sm[2], sm[3]));
        float L = 0.f; float4 o = {0.f, 0.f, 0.f, 0.f};
        for (int w2 = 0; w2 < 4; ++w2) {
            const float c = __expf(sm[w2] - M);
            L += sl[w2] * c;
            o.x += sacc[w2][lane * 4 + 0] * c;
            o.y += sacc[w2][lane * 4 + 1] * c;
            o.z += sacc[w2][lane * 4 + 2] * c;
            o.w += sacc[w2][lane * 4 + 3] * c;
        }
        const float iL = 1.f / L;
        float4 r = {o.x * iL, o.y * iL, o.z * iL, o.w * iL};
        *(float4*)(attn_out + (size_t)b * (NH_ * DV_) + h * DV_ + lane * 4) = r;
    }
}

// =====================================================================
// Router: logits -> softmax -> top-4 -> renormalized cw[B,E]
// =====================================================================
__global__ __launch_bounds__(256)
void router_kernel(const float* __restrict__ h2, const float* __restrict__ wg,
                   float* __restrict__ cw)
{
    const int b = blockIdx.x;
    __shared__ float part[8][E_];
    const int e = threadIdx.x & 31, w = threadIdx.x >> 5;
    float s = 0.f;
    for (int hh = w; hh < HD_; hh += 8)
        s += h2[(size_t)b * HD_ + hh] * wg[(size_t)hh * E_ + e];
    part[w][e] = s;
    __syncthreads();
    if (threadIdx.x < 32) {
        float logit = 0.f;
        for (int ww = 0; ww < 8; ++ww) logit += part[ww][e];
        float mx = logit;
        for (int off = 16; off; off >>= 1) mx = fmaxf(mx, __shfl_xor(mx, off, 32));
        float p = __expf(logit - mx);
        float sum = p;
        for (int off = 16; off; off >>= 1) sum += __shfl_xor(sum, off, 32);
        p /= sum;
        float rem = p, tsum = 0.f; int sel = 0;
        for (int t = 0; t < TOPK_; ++t) {
            float v = rem; int idx = e;
            for (int off = 16; off; off >>= 1) {
                float ov = __shfl_xor(v, off, 32);
                int   oi = __shfl_xor(idx, off, 32);
                if (ov > v || (ov == v && oi < idx)) { v = ov; idx = oi; }
            }
            if (e == idx) { sel = 1; rem = -1.f; }
            tsum += v;
        }
        cw[(size_t)b * E_ + e] = sel ? (p / tsum) : 0.f;
    }
}

// =====================================================================
// MoE up+gate: gu = h2 @ w1[e]^T (w1: [E, 2I, HD], NT)
//   act[e,b,i] = silu(g)*u * cw[b,e];  grid = (I/64, E)
// =====================================================================
__global__ __launch_bounds__(256)
void moe_act_kernel(const float* __restrict__ h2, const float* __restrict__ w1,
                    const float* __restrict__ cw, float* __restrict__ act)
{
    __shared__ __align__(16) float As[MT * AP];
    __shared__ __align__(16) float BgT[NT * AP];
    __shared__ __align__(16) float BuT[NT * AP];
    const int n0 = blockIdx.x * NT;
    const int e  = blockIdx.y;
    const int wave = threadIdx.x >> 5;
    const int u0 = wave, u1 = wave + 8;
    const float* w1g = w1 + (size_t)e * 2 * I_ * HD_;   // rows [0, I)
    const float* w1u = w1g + (size_t)I_ * HD_;          // rows [I, 2I)
    v8f g0 = {}, g1 = {}, uu0 = {}, uu1 = {};

    for (int k0 = 0; k0 < HD_; k0 += KC) {
        __syncthreads();
        loadA(As, h2, HD_, k0);
        loadB_nt(BgT, w1g, HD_, n0, k0);
        loadB_nt(BuT, w1u, HD_, n0, k0);
        cp_wait();
        __syncthreads();
        g0  = mma_nt(As, BgT, u0 >> 2, u0 & 3, g0);
        g1  = mma_nt(As, BgT, u1 >> 2, u1 & 3, g1);
        uu0 = mma_nt(As, BuT, u0 >> 2, u0 & 3, uu0);
        uu1 = mma_nt(As, BuT, u1 >> 2, u1 & 3, uu1);
    }
    const int lane = threadIdx.x & 31, half = lane >> 4, l = lane & 15;
    float* aout = act + (size_t)e * B_ * I_;
#pragma unroll
    for (int part = 0; part < 2; ++part) {
        const int u = part ? u1 : u0;
        v8f g  = part ? g1 : g0;
        v8f uv = part ? uu1 : uu0;
        const int n = n0 + (u & 3) * 16 + l;
#pragma unroll
        for (int r = 0; r < 8; ++r) {
            const int mrow = (u >> 2) * 16 + half * 8 + r;   // token index
            const float gg = g[r];
            const float silu = gg / (1.f + __expf(-gg));
            aout[(size_t)mrow * I_ + n] = silu * uv[r] * cw[(size_t)mrow * E_ + e];
        }
    }
}

// =====================================================================
// Shared-expert up+gate: gs_us = h2 @ wse (wse [HD, 2I], NN);  grid = I/64
// =====================================================================
__global__ __launch_bounds__(256)
void shared_act_kernel(const float* __restrict__ h2, const float* __restrict__ wse,
                       float* __restrict__ act_s)
{
    __shared__ __align__(16) float As[MT * AP];
    __shared__ __align__(16) float Bg[KC * BPN];
    __shared__ __align__(16) float Bu[KC * BPN];
    const int n0 = blockIdx.x * NT;
    const int wave = threadIdx.x >> 5;
    const int u0 = wave, u1 = wave + 8;
    v8f g0 = {}, g1 = {}, uu0 = {}, uu1 = {};

    for (int k0 = 0; k0 < HD_; k0 += KC) {
        __syncthreads();
        loadA(As, h2, HD_, k0);
        loadB_nn(Bg, wse, 2 * I_, n0, k0);
        loadB_nn(Bu, wse, 2 * I_, I_ + n0, k0);
        cp_wait();
        __syncthreads();
        g0  = mma_nn(As, Bg, u0 >> 2, u0 & 3, g0);
        g1  = mma_nn(As, Bg, u1 >> 2, u1 & 3, g1);
        uu0 = mma_nn(As, Bu, u0 >> 2, u0 & 3, uu0);
        uu1 = mma_nn(As, Bu, u1 >> 2, u1 & 3, uu1);
    }
    const int lane = threadIdx.x & 31, half = lane >> 4, l = lane & 15;
#pragma unroll
    for (int part = 0; part < 2; ++part) {
        const int u = part ? u1 : u0;
        v8f g  = part ? g1 : g0;
        v8f uv = part ? uu1 : uu0;
        const int n = n0 + (u & 3) * 16 + l;
#pragma unroll
        for (int r = 0; r < 8; ++r) {
            const int mrow = (u >> 2) * 16 + half * 8 + r;
            const float gg = g[r];
            act_s[(size_t)mrow * I_ + n] = (gg / (1.f + __expf(-gg))) * uv[r];
        }
    }
}

// =====================================================================
// MoE down + final combine:
//   out[b,n] = x1[b,n] + shared[b,n] + sum_e act[e,b,:] @ w2[e][n,:]  (NT)
// =====================================================================
__global__ __launch_bounds__(256)
void moe_down_kernel(const float* __restrict__ act, const float* __restrict__ w2,
                     const float* __restrict__ x1, const float* __restrict__ shared_o,
                     float* __restrict__ out)
{
    __shared__ __align__(16) float As[MT * AP];
    __shared__ __align__(16) float BT[NT * AP];
    const int n0 = blockIdx.x * NT;
    const int wave = threadIdx.x >> 5;
    const int u0 = wave, u1 = wave + 8;
    v8f acc0 = {}, acc1 = {};

    for (int e = 0; e < E_; ++e) {
        const float* ae = act + (size_t)e * B_ * I_;
        const float* we = w2 + (size_t)e * HD_ * I_;
        for (int k0 = 0; k0 < I_; k0 += KC) {
            __syncthreads();
            loadA(As, ae, I_, k0);
            loadB_nt(BT, we, I_, n0, k0);
            cp_wait();
            __syncthreads();
            acc0 = mma_nt(As, BT, u0 >> 2, u0 & 3, acc0);
            acc1 = mma_nt(As, BT, u1 >> 2, u1 & 3, acc1);
        }
    }
    const int lane = threadIdx.x & 31, half = lane >> 4, l = lane & 15;
#pragma unroll
    for (int part = 0; part < 2; ++part) {
        const int u = part ? u1 : u0;
        v8f a = part ? acc1 : acc0;
        const int n = n0 + (u & 3) * 16 + l;
#pragma unroll
        for (int r = 0; r < 8; ++r) {
            const int m = (u >> 2) * 16 + half * 8 + r;
            out[(size_t)m * HD_ + n] =
                x1[(size_t)m * HD_ + n] + shared_o[(size_t)m * HD_ + n] + a[r];
        }
    }
}

// =====================================================================
// Launcher
// =====================================================================
extern "C" void kernel_launch(void* const* d_in, const int* in_sizes, int n_in,
                              void* d_out, int out_size, void* d_ws, size_t ws_size,
                              hipStream_t stream)
{
    const float* x       = (const float*)d_in[0];
    const float* k_cache = (const float*)d_in[1];
    const float* v_cache = (const float*)d_in[2];
    const float* w_ln1   = (const float*)d_in[3];
    const float* w_ln2   = (const float*)d_in[4];
    const float* wq      = (const float*)d_in[5];
    const float* wkv     = (const float*)d_in[6];
    const float* wo      = (const float*)d_in[7];
    const float* wg      = (const float*)d_in[8];
    const float* w1      = (const float*)d_in[9];
    const float* w2      = (const float*)d_in[10];
    const float* wse     = (const float*)d_in[11];
    const float* wsd     = (const float*)d_in[12];
    const int* slot_map  = (const int*)d_in[13];
    const int* seq_lens  = (const int*)d_in[14];
    float* out = (float*)d_out;

    const int NQ  = NH_ * QHD_;                    // 3072
    const int NKV = NH_ * (DKN_ + DV_) + DKR_;     // 4160

    float* ws = (float*)d_ws;
    float* h        = ws;  ws += (size_t)B_ * HD_;
    float* qf       = ws;  ws += (size_t)B_ * NQ;
    float* kvf      = ws;  ws += (size_t)B_ * NKV;
    float* attn_o   = ws;  ws += (size_t)B_ * HD_;
    float* x1       = ws;  ws += (size_t)B_ * HD_;
    float* h2       = ws;  ws += (size_t)B_ * HD_;
    float* cw       = ws;  ws += (size_t)B_ * E_;
    float* act      = ws;  ws += (size_t)E_ * B_ * I_;
    float* act_s    = ws;  ws += (size_t)B_ * I_;
    float* shared_o = ws;  ws += (size_t)B_ * HD_;

    rmsnorm_kernel<<<B_, 256, 0, stream>>>(x, w_ln1, h);
    gemm64_kernel<<<NQ / NT, 256, 0, stream>>>(h, HD_, wq, NQ, qf, NQ,
                                               nullptr, 0, HD_);
    gemm64_kernel<<<NKV / NT, 256, 0, stream>>>(h, HD_, wkv, NKV, kvf, NKV,
                                                nullptr, 0, HD_);
    dim3 ag(NH_, B_);
    attn_kernel<<<ag, 128, 0, stream>>>(qf, kvf, k_cache, v_cache,
                                        slot_map, seq_lens, attn_o);
    gemm64_kernel<<<HD_ / NT, 256, 0, stream>>>(attn_o, HD_, wo, HD_, x1, HD_,
                                                x, HD_, HD_);
    rmsnorm_kernel<<<B_, 256, 0, stream>>>(x1, w_ln2, h2);
    router_kernel<<<B_, 256, 0, stream>>>(h2, wg, cw);
    dim3 mg(I_ / NT, E_);
    moe_act_kernel<<<mg, 256, 0, stream>>>(h2, w1, cw, act);
    shared_act_kernel<<<I_ / NT, 256, 0, stream>>>(h2, wse, act_s);
    gemm64_kernel<<<HD_ / NT, 256, 0, stream>>>(act_s, I_, wsd, HD_, shared_o,
                                                HD_, nullptr, 0, I_);
    moe_down_kernel<<<HD_ / NT, 256, 0, stream>>>(act, w2, x1, shared_o, out);
}